// MulLayer_73976516706890
// MI455X (gfx1250) — compile-verified
//
#include <hip/hip_runtime.h>
#include <stdint.h>

// Graph message-passing mean-aggregation (DGL-style) for MI455X / gfx1250.
//
// Roofline: ~0.12 GFLOP vs ~1 GB of traffic -> pure bandwidth problem
// (~45us floor at 23.3 TB/s). No dense matmul exists in this op, so WMMA is
// intentionally NOT used. CDNA5-specific paths used instead:
//   * global_load_async_to_lds_b128 (ASYNCcnt), double-buffered 2-deep so the
//     streamed edge row for e+stride is in flight while e is consumed
//   * th:TH_LOAD_NT on the edge stream (read-once, 327MB) to keep the 51MB
//     atomic-scatter target resident in the 192MB L2
//   * ds_load_b128 readback; s_wait_asynccnt 1 / s_wait_dscnt 0 split counters
//   * global_atomic_add_f32 scatter (no-return, STOREcnt, L2-rate)
//   * global_prefetch_b8 for the next gathered src row
//   * wave32 tiling: 1 wave (32 lanes x 16B) == one 512B embedding row

#define DIMS 128
#define ALPHA 0.3f
#define WAVES_PER_BLOCK 8

typedef float v4f __attribute__((ext_vector_type(4)));
typedef __attribute__((address_space(3))) float lds_float;

// ---------------------------------------------------------------- init ----
__global__ void gnn_init_kernel(float* __restrict__ out, float* __restrict__ deg,
                                int n_out4, int n_nodes) {
  int tid = blockIdx.x * blockDim.x + threadIdx.x;
  int stride = gridDim.x * blockDim.x;
  v4f z = {0.f, 0.f, 0.f, 0.f};
  for (int i = tid; i < n_out4; i += stride)
    reinterpret_cast<v4f*>(out)[i] = z;
  for (int i = tid; i < n_nodes; i += stride)
    deg[i] = 0.f;
}

// ------------------------------------------------------------- scatter ----
// One wave per edge (grid-stride). Lane l owns channels [4l, 4l+4).
// Edge rows are streamed global->LDS with the async engine, double-buffered.

__device__ __forceinline__ void async_fill_row(unsigned ldsOff,
                                               const float* rowBase,
                                               int lane) {
  uint64_t gaddr = (uint64_t)(uintptr_t)rowBase + (uint64_t)(lane * 16);
  // Non-temporal: each edge row is touched exactly once.
  asm volatile("global_load_async_to_lds_b128 %0, %1, off th:TH_LOAD_NT"
               :: "v"(ldsOff), "v"(gaddr)
               : "memory");
}

__global__ __launch_bounds__(WAVES_PER_BLOCK * 32)
void gnn_scatter_kernel(const float* __restrict__ src_emb,
                        const float* __restrict__ edge_emb,
                        const int* __restrict__ src_idx,
                        const int* __restrict__ dst_idx,
                        float* __restrict__ out,
                        float* __restrict__ deg,
                        int n_edges) {
  // 2 buffers per wave -> 8KB of the 320KB WGP LDS pool.
  __shared__ float tile[WAVES_PER_BLOCK * 2 * DIMS];

  const int lane = threadIdx.x & 31;
  const int wib  = threadIdx.x >> 5;               // wave index in block
  const int wave = blockIdx.x * WAVES_PER_BLOCK + wib;
  const int wstride = gridDim.x * WAVES_PER_BLOCK;

  // LDS byte offsets of this lane's 16B slice in each buffer, derived from
  // the real AS(3) address of `tile` so async writes and ds_loads agree.
  const unsigned ldsBase = (unsigned)(size_t)(lds_float*)tile;
  const unsigned off0 =
      ldsBase + (unsigned)(((wib * 2 + 0) * DIMS + lane * 4) * sizeof(float));
  const unsigned off1 =
      ldsBase + (unsigned)(((wib * 2 + 1) * DIMS + lane * 4) * sizeof(float));

  int e = wave;
  if (e >= n_edges) return;

  // Prologue: start the fill for the first edge into buffer 0.
  async_fill_row(off0, edge_emb + (size_t)e * DIMS, lane);

  int buf = 0;
  for (; e < n_edges; e += wstride) {
    const int en = e + wstride;
    // Keep exactly 2 async fills in flight at the wait point: on the last
    // iteration re-issue the current row (harmless) into the spare buffer.
    const int epre = (en < n_edges) ? en : e;
    async_fill_row(buf ? off0 : off1, edge_emb + (size_t)epre * DIMS, lane);

    const int s  = src_idx[e];
    const int dn = dst_idx[e];

    // Prefetch the NEXT iteration's gathered src row (global_prefetch_b8)
    // while both async fills and the current gather are in flight.
    if (en < n_edges) {
      int s2 = src_idx[en];
      __builtin_prefetch(src_emb + (size_t)s2 * DIMS + lane * 4, 0, 0);
    }

    // Random-access gather of the src row (b128 per lane, default RT:
    // each src row is reused ~E/N = 6.4x across the kernel).
    v4f sv = *reinterpret_cast<const v4f*>(src_emb + (size_t)s * DIMS + lane * 4);

    // Async loads complete in order: cnt<=1 => the older fill (edge e) done.
    asm volatile("s_wait_asynccnt 1" ::: "memory");
    v4f ev;
    asm volatile("ds_load_b128 %0, %1\n\t"
                 "s_wait_dscnt 0"
                 : "=v"(ev)
                 : "v"(buf ? off1 : off0)
                 : "memory");

    v4f m = sv + ev;                               // msg[e] slice

    // Scatter: f32 atomics into d_out (51MB -> resident in 192MB L2).
    float* op = out + (size_t)dn * DIMS + lane * 4;
    atomicAdd(op + 0, m.x);
    atomicAdd(op + 1, m.y);
    atomicAdd(op + 2, m.z);
    atomicAdd(op + 3, m.w);
    if (lane == 0) atomicAdd(deg + dn, 1.0f);      // degree count

    buf ^= 1;
  }
}

// ------------------------------------------------------------ finalize ----
// out[n,:] = deg>0 ? ALPHA*dst + (1-ALPHA)*sum/deg : 0
__global__ void gnn_finalize_kernel(const float* __restrict__ dst_emb,
                                    const float* __restrict__ deg,
                                    float* __restrict__ out,
                                    int n_nodes) {
  int tid = blockIdx.x * blockDim.x + threadIdx.x;
  int stride = gridDim.x * blockDim.x;
  int total = n_nodes * (DIMS / 4);
  for (int i = tid; i < total; i += stride) {
    int n = i >> 5;                                // i / (DIMS/4)
    float dg = deg[n];
    v4f sum = reinterpret_cast<v4f*>(out)[i];
    v4f dv  = reinterpret_cast<const v4f*>(dst_emb)[i];
    v4f r;
    if (dg > 0.f) {
      float w = (1.0f - ALPHA) / dg;
      r = ALPHA * dv + w * sum;
    } else {
      r = (v4f){0.f, 0.f, 0.f, 0.f};
    }
    reinterpret_cast<v4f*>(out)[i] = r;
  }
}

// -------------------------------------------------------------- launch ----
extern "C" void kernel_launch(void* const* d_in, const int* in_sizes, int n_in,
                              void* d_out, int out_size, void* d_ws, size_t ws_size,
                              hipStream_t stream) {
  const float* src_emb  = (const float*)d_in[0];   // [N, 128]
  const float* dst_emb  = (const float*)d_in[1];   // [N, 128]
  const float* edge_emb = (const float*)d_in[2];   // [E, 128]
  const int*   src_idx  = (const int*)d_in[3];     // [E]
  const int*   dst_idx  = (const int*)d_in[4];     // [E]
  float* out = (float*)d_out;                      // [N, 128]
  float* deg = (float*)d_ws;                       // [N] floats of scratch

  const int n_nodes = in_sizes[0] / DIMS;
  const int n_edges = in_sizes[3];
  const int n_out4  = n_nodes * (DIMS / 4);

  gnn_init_kernel<<<2048, 256, 0, stream>>>(out, deg, n_out4, n_nodes);
  gnn_scatter_kernel<<<4096, WAVES_PER_BLOCK * 32, 0, stream>>>(
      src_emb, edge_emb, src_idx, dst_idx, out, deg, n_edges);
  gnn_finalize_kernel<<<4096, 256, 0, stream>>>(dst_emb, deg, out, n_nodes);
}